// CircularUpsample2_76089640616433
// MI455X (gfx1250) — compile-verified
//
#include <hip/hip_runtime.h>
#include <stdint.h>

typedef __attribute__((ext_vector_type(4))) float v4f;

#define IMG_IN  (128 * 128)
#define IMG_OUT (256 * 256)

// Circular 2x upsample with FIR kernel outer([1,3,3,1])/16 * 4.
// Polyphase form: out(2iy+py, 2ix+px) = sum over 2x2 input patch at
// rows {(iy-2)&127,(iy-1)&127}, cols {(ix-2)&127,(ix-1)&127} with
// per-dim weights (3,1) for even parity, (1,3) for odd, scaled 1/16.
__global__ __launch_bounds__(128)
void upsample2_circ_kernel(const float* __restrict__ x, float* __restrict__ out) {
    __shared__ float sm[3 * 128];

    const int blk = blockIdx.x;
    const int bc  = blk >> 6;          // image index (b*c), 0..2047
    const int iy0 = (blk & 63) << 1;   // first of 2 input-row indices handled

    const int t = threadIdx.x;         // 0..127

    const float* img = x + (size_t)bc * IMG_IN;

    // ---- async stage: 3 input rows (iy0-2, iy0-1, iy0) mod 128 -> LDS ----
    // 96 lanes (waves 0..2), each moves 16 bytes via async b128 to LDS.
    if (t < 96) {
        const int r  = t >> 5;                 // lds row 0..2
        const int c  = (t & 31) << 2;          // float column (0,4,...,124)
        const int gy = (iy0 - 2 + r) & 127;    // wrapped source row
        unsigned lds_addr = (unsigned)(uintptr_t)(&sm[r * 128 + c]);
        int      goff     = (gy * 128 + c) * 4;   // byte offset from image base
        asm volatile("global_load_async_to_lds_b128 %0, %1, %2 offset:0"
                     :: "v"(lds_addr), "v"(goff),
                        "s"((uint64_t)(uintptr_t)img)
                     : "memory");
    }
    asm volatile("s_wait_asynccnt 0x0" ::: "memory");
    __syncthreads();

    // ---- compute: thread -> (row-pair s, column quad q) ----
    const int s  = t >> 6;     // which input row of the pair (0/1)
    const int q  = t & 63;     // covers input cols 2q, 2q+1 -> output cols 4q..4q+3
    const int iy = iy0 + s;

    const float* r0 = &sm[s * 128];        // input row (iy-2)&127
    const float* r1 = &sm[(s + 1) * 128];  // input row (iy-1)&127

    const int xa = (2 * q - 2) & 127;
    const int xb = (2 * q - 1) & 127;
    const int xc = (2 * q)     & 127;

    const float a0 = r0[xa], a1 = r0[xb], a2 = r0[xc];
    const float c0 = r1[xa], c1 = r1[xb], c2 = r1[xc];

    const float S = 0.0625f;  // 1/16
    v4f top, bot;
    // output row 2*iy (even y: weights (3,1) on rows r0,r1)
    top.x = (9.f * a0 + 3.f * a1 + 3.f * c0 +       c1) * S;  // even x
    top.y = (3.f * a0 + 9.f * a1 +       c0 + 3.f * c1) * S;  // odd  x
    top.z = (9.f * a1 + 3.f * a2 + 3.f * c1 +       c2) * S;
    top.w = (3.f * a1 + 9.f * a2 +       c1 + 3.f * c2) * S;
    // output row 2*iy+1 (odd y: weights (1,3) on rows r0,r1)
    bot.x = (3.f * a0 +       a1 + 9.f * c0 + 3.f * c1) * S;
    bot.y = (      a0 + 3.f * a1 + 3.f * c0 + 9.f * c1) * S;
    bot.z = (3.f * a1 +       a2 + 9.f * c1 + 3.f * c2) * S;
    bot.w = (      a1 + 3.f * a2 + 3.f * c1 + 9.f * c2) * S;

    float* op = out + (size_t)bc * IMG_OUT + (size_t)(2 * iy) * 256 + 4 * q;
    __builtin_nontemporal_store(top, (v4f*)op);          // row 2*iy
    __builtin_nontemporal_store(bot, (v4f*)(op + 256));  // row 2*iy+1
}

extern "C" void kernel_launch(void* const* d_in, const int* in_sizes, int n_in,
                              void* d_out, int out_size, void* d_ws, size_t ws_size,
                              hipStream_t stream) {
    const float* x = (const float*)d_in[0];
    // d_in[1] is the fixed FIR kernel outer([1,3,3,1])/16 * 4; weights are
    // baked into the kernel above (exact in fp32: 9/16, 3/16, 1/16).
    float* out = (float*)d_out;

    const int n_img = in_sizes[0] / IMG_IN;  // b*c images (8*256 = 2048)
    dim3 grid(n_img * 64);                   // 64 input-row-pairs per image
    dim3 block(128);
    hipLaunchKernelGGL(upsample2_circ_kernel, grid, block, 0, stream, x, out);
}